// StackedAttention_41051297415636
// MI455X (gfx1250) — compile-verified
//
#include <hip/hip_runtime.h>
#include <hip/hip_bf16.h>

// Fixed problem shape from the reference.
#define BB 4
#define CC 256
#define NN 1024
#define C4 64
#define PH 16
#define EPS_BN 1e-5f
#define KSTEP 32   // K-panel depth staged in LDS for the attention GEMM

typedef __attribute__((ext_vector_type(2))) float v2f;
typedef __attribute__((ext_vector_type(8))) float v8f;

__device__ __forceinline__ v8f wmma_f32(v2f a, v2f b, v8f c) {
  // V_WMMA_F32_16X16X4_F32 : D(16x16,f32) = A(16x4,f32) * B(4x16,f32) + C
  return __builtin_amdgcn_wmma_f32_16x16x4_f32(false, a, false, b, (short)0, c,
                                               false, false);
}

// ---- CDNA5 async global->LDS path (guarded; fallback = sync copy) ----------
#if defined(__HIP_DEVICE_COMPILE__) && \
    __has_builtin(__builtin_amdgcn_global_load_async_to_lds_b128)
#define USE_ASYNC_LDS 1
#else
#define USE_ASYNC_LDS 0
#endif

#if USE_ASYNC_LDS
// Builtin signature (from hipcc diagnostic): first parameter is
// 'int __vector(4) __device__ *' i.e. a 16-byte vector pointer in AS1.
typedef int v4i __attribute__((vector_size(16)));
typedef __attribute__((address_space(1))) v4i as1_v4i;
typedef __attribute__((address_space(3))) v4i as3_v4i;
#if __has_builtin(__builtin_amdgcn_s_wait_asynccnt)
#define WAIT_ASYNC() __builtin_amdgcn_s_wait_asynccnt(0)
#else
#define WAIT_ASYNC() asm volatile("s_wait_asynccnt 0x0" ::: "memory")
#endif
#else
#define WAIT_ASYNC()
#endif

// Copy 16 bytes global -> LDS (per lane).
__device__ __forceinline__ void copy16_g2l(const float* g, float* l) {
#if USE_ASYNC_LDS
  __builtin_amdgcn_global_load_async_to_lds_b128((as1_v4i*)g, (as3_v4i*)l, 0, 0);
#else
  *(float4*)l = *(const float4*)g;
#endif
}

// ---------------------------------------------------------------------------
// Generic GEMM: Y[b,o,n] = sum_c W[o,c] * X[b,c,n] (+ bias[o] if non-null)
// grid = (NN/16, O/16, BB), block = 32 (one wave per 16x16 tile).
// A fragment layout (documented 32-bit 16x4): lanes 0-15 M=0..15 hold K={k,k+1}
// in {v0,v1}; lanes 16-31 hold K={k+2,k+3}. B fragment uses the symmetric
// split for its 4x16 rows.
// ---------------------------------------------------------------------------
__global__ void k_gemm(const float* __restrict__ W, const float* __restrict__ X,
                       const float* __restrict__ bias, float* __restrict__ Y,
                       int K) {
  const int lane = threadIdx.x;
  const int half = lane >> 4;
  const int l16  = lane & 15;
  const int b  = blockIdx.z;
  const int o0 = blockIdx.y * 16;
  const int n0 = blockIdx.x * 16;
  const int O  = gridDim.y * 16;
  const float* Xb = X + (size_t)b * K * NN;
  v8f acc = {};
#pragma unroll 4
  for (int k = 0; k < K; k += 4) {
    const int ka = k + half * 2;
    v2f a, bm;
    a.x  = W[(size_t)(o0 + l16) * K + ka];
    a.y  = W[(size_t)(o0 + l16) * K + ka + 1];
    bm.x = Xb[(size_t)ka * NN + n0 + l16];
    bm.y = Xb[(size_t)(ka + 1) * NN + n0 + l16];
    acc = wmma_f32(a, bm, acc);
  }
  float* Yb = Y + (size_t)b * O * NN;
  const int n = n0 + l16;
#pragma unroll
  for (int r = 0; r < 8; ++r) {
    const int m = o0 + r + half * 8;
    float v = acc[r];
    if (bias) v += bias[m];
    Yb[(size_t)m * NN + n] = v;
  }
}

// ---------------------------------------------------------------------------
// BatchNorm training stats: per-channel mean and biased variance over (B,N).
// grid = CC, block = 256.
// ---------------------------------------------------------------------------
__global__ void k_bn_stats(const float* __restrict__ T, float* __restrict__ mean,
                           float* __restrict__ var) {
  const int c = blockIdx.x;
  float s = 0.f, s2 = 0.f;
  for (int i = threadIdx.x; i < BB * NN; i += 256) {
    const int bb = i >> 10;          // / NN
    const int n  = i & (NN - 1);
    const float v = T[((size_t)bb * CC + c) * NN + n];
    s  += v;
    s2 += v * v;
  }
  __shared__ float sh[256], sh2[256];
  sh[threadIdx.x] = s; sh2[threadIdx.x] = s2;
  __syncthreads();
  for (int off = 128; off > 0; off >>= 1) {
    if (threadIdx.x < off) {
      sh[threadIdx.x]  += sh[threadIdx.x + off];
      sh2[threadIdx.x] += sh2[threadIdx.x + off];
    }
    __syncthreads();
  }
  if (threadIdx.x == 0) {
    const float m = sh[0] / (float)(BB * NN);
    mean[c] = m;
    var[c]  = sh2[0] / (float)(BB * NN) - m * m;
  }
}

// Y = relu(gamma*(T-mu)*rsqrt(var+eps)+beta), elementwise over (B,C,N).
__global__ void k_bn_relu(const float* __restrict__ T, const float* __restrict__ gamma,
                          const float* __restrict__ beta, const float* __restrict__ mean,
                          const float* __restrict__ var, float* __restrict__ Y) {
  const size_t i = (size_t)blockIdx.x * blockDim.x + threadIdx.x;
  const int c = (int)((i >> 10) & (CC - 1));
  const float v = gamma[c] * (T[i] - mean[c]) * rsqrtf(var[c] + EPS_BN) + beta[c];
  Y[i] = fmaxf(v, 0.f);
}

// P[b,n,h] = sum_i xyz[b,n,i] * W1[i,h]   (W1 is (3,PH) row-major)
__global__ void k_pos_embed(const float* __restrict__ xyz, const float* __restrict__ W1,
                            float* __restrict__ P) {
  const int i  = blockIdx.x * blockDim.x + threadIdx.x;  // b*N*16 + n*16 + h
  const int h  = i & 15;
  const int bn = i >> 4;
  const float* p = xyz + (size_t)bn * 3;
  P[i] = p[0] * W1[0 * PH + h] + p[1] * W1[1 * PH + h] + p[2] * W1[2 * PH + h];
}

// ---------------------------------------------------------------------------
// Scores: S[b,n,m] = sum_d Q[b,d,n]*Q[b,d,m]
//                  + sum_h relu(P[b,n,h]-P[b,m,h]+b1[h])*W2[h] + b2
// Gram via WMMA (K=64); positional bias fused in the epilogue.
// grid = (NN/16 [m], NN/16 [n], BB), block = 32.
// ---------------------------------------------------------------------------
__global__ void k_energy(const float* __restrict__ Q, const float* __restrict__ P,
                         const float* __restrict__ b1, const float* __restrict__ W2,
                         const float* __restrict__ b2, float* __restrict__ S) {
  const int lane = threadIdx.x;
  const int half = lane >> 4;
  const int l16  = lane & 15;
  const int b  = blockIdx.z;
  const int n0 = blockIdx.y * 16;
  const int m0 = blockIdx.x * 16;
  const float* Qb = Q + (size_t)b * C4 * NN;
  v8f acc = {};
#pragma unroll 4
  for (int d = 0; d < C4; d += 4) {
    const int ka = d + half * 2;
    v2f a, bm;
    a.x  = Qb[(size_t)ka * NN + n0 + l16];
    a.y  = Qb[(size_t)(ka + 1) * NN + n0 + l16];
    bm.x = Qb[(size_t)ka * NN + m0 + l16];
    bm.y = Qb[(size_t)(ka + 1) * NN + m0 + l16];
    acc = wmma_f32(a, bm, acc);
  }
  const float* Pb = P + (size_t)b * NN * PH;
  float pm[PH];
#pragma unroll
  for (int h = 0; h < PH; ++h) pm[h] = Pb[(size_t)(m0 + l16) * PH + h];
  const float bias2 = b2[0];
  float* Sb = S + (size_t)b * NN * NN;
#pragma unroll
  for (int r = 0; r < 8; ++r) {
    const int n = n0 + r + half * 8;
    const float* pn = Pb + (size_t)n * PH;
    float pb = bias2;
#pragma unroll
    for (int h = 0; h < PH; ++h) {
      const float t = pn[h] - pm[h] + b1[h];
      pb += fmaxf(t, 0.f) * W2[h];
    }
    Sb[(size_t)n * NN + m0 + l16] = acc[r] + pb;
  }
}

// Row softmax over m (in place). grid = B*N rows, block = 256.
__global__ void k_softmax(float* __restrict__ S) {
  float* r = S + (size_t)blockIdx.x * NN;
  const int t = threadIdx.x;
  __shared__ float sh[256];
  float mx = -3.402823e38f;
  for (int j = t; j < NN; j += 256) mx = fmaxf(mx, r[j]);
  sh[t] = mx; __syncthreads();
  for (int off = 128; off > 0; off >>= 1) {
    if (t < off) sh[t] = fmaxf(sh[t], sh[t + off]);
    __syncthreads();
  }
  mx = sh[0]; __syncthreads();
  float s = 0.f;
  for (int j = t; j < NN; j += 256) {
    const float e = __expf(r[j] - mx);
    r[j] = e;
    s += e;
  }
  sh[t] = s; __syncthreads();
  for (int off = 128; off > 0; off >>= 1) {
    if (t < off) sh[t] += sh[t + off];
    __syncthreads();
  }
  const float invs = 1.0f / sh[0];
  for (int j = t; j < NN; j += 256) r[j] *= invs;
}

// inv[b,m] = 1 / (1e-9 + sum_n attn[b,n,m]). grid = B*N/256, block = 256.
__global__ void k_colsum_inv(const float* __restrict__ S, float* __restrict__ inv) {
  const int g = blockIdx.x * blockDim.x + threadIdx.x;   // b*N + m
  const int b = g >> 10;
  const int m = g & (NN - 1);
  const float* Sb = S + (size_t)b * NN * NN + m;
  float s = 0.f;
  for (int n = 0; n < NN; ++n) s += Sb[(size_t)n * NN];
  inv[g] = 1.0f / (1e-9f + s);
}

// ---------------------------------------------------------------------------
// U[b,c,m] = X[b,c,m] - inv[b,m] * sum_n XV[b,c,n] * A[b,n,m]   (WMMA, K=1024)
// Block = 128 threads (4 waves) computing a 64(c) x 16(m) tile. The 4 waves
// share a KSTEP x 16 panel of A (attn) staged in LDS via async global->LDS
// copies (double buffered); each wave owns one 16-row c-tile of XV.
// grid = (NN/16 [m], CC/64 [c], BB).
// ---------------------------------------------------------------------------
__global__ __launch_bounds__(128)
void k_xr_sub(const float* __restrict__ XV, const float* __restrict__ A,
              const float* __restrict__ invc, const float* __restrict__ X,
              float* __restrict__ U) {
  const int tid  = threadIdx.x;
  const int wave = tid >> 5;       // 0..3
  const int lane = tid & 31;
  const int half = lane >> 4;
  const int l16  = lane & 15;
  const int b  = blockIdx.z;
  const int m0 = blockIdx.x * 16;
  const int c0 = blockIdx.y * 64 + wave * 16;

  const float* XVrow = XV + (size_t)b * CC * NN + (size_t)(c0 + l16) * NN;
  const float* Ab    = A + (size_t)b * NN * NN;

  __shared__ float panel[2][KSTEP * 16];   // 2 x 2KB double buffer

  // Stage rows [wave*8, wave*8+8) of the K-panel starting at k0 into buffer bufi.
  // Each lane moves 16B: row kk = wave*8 + lane/4, cols m0+(lane%4)*4 .. +3.
  const int kk = wave * 8 + (lane >> 2);
  const int mo = (lane & 3) * 4;

#define STAGE(k0, bufi)                                                     \
  copy16_g2l(Ab + (size_t)((k0) + kk) * NN + m0 + mo,                       \
             &panel[bufi][kk * 16 + mo])

  STAGE(0, 0);
  WAIT_ASYNC();
  __syncthreads();

  v8f acc = {};
  const int nkb = NN / KSTEP;              // 32 panels
  for (int kb = 0; kb < nkb; ++kb) {
    const int bufi = kb & 1;
    if (kb + 1 < nkb) STAGE((kb + 1) * KSTEP, bufi ^ 1);
    const float* p = panel[bufi];
    const int kbase = kb * KSTEP;
#pragma unroll
    for (int ks = 0; ks < KSTEP; ks += 4) {
      const int ka = ks + half * 2;
      v2f a, bm;
      a.x  = XVrow[kbase + ka];
      a.y  = XVrow[kbase + ka + 1];
      bm.x = p[ka * 16 + l16];
      bm.y = p[(ka + 1) * 16 + l16];
      acc = wmma_f32(a, bm, acc);
    }
    WAIT_ASYNC();
    __syncthreads();
  }
#undef STAGE

  const float iv = invc[b * NN + m0 + l16];
  const float* Xb = X + (size_t)b * CC * NN;
  float* Ub = U + (size_t)b * CC * NN;
  const int m = m0 + l16;
#pragma unroll
  for (int r = 0; r < 8; ++r) {
    const int c = c0 + r + half * 8;
    Ub[(size_t)c * NN + m] = Xb[(size_t)c * NN + m] - acc[r] * iv;
  }
}

// x_new = x + relu(BN(t)); written to ping-pong buffer and output slice.
__global__ void k_residual(const float* __restrict__ X, const float* __restrict__ T,
                           const float* __restrict__ mean, const float* __restrict__ var,
                           const float* __restrict__ gamma, const float* __restrict__ beta,
                           float* __restrict__ XN, float* __restrict__ Out) {
  const size_t i = (size_t)blockIdx.x * blockDim.x + threadIdx.x;   // b*C*N range
  const int c = (int)((i >> 10) & (CC - 1));
  const int b = (int)(i / ((size_t)CC * NN));
  const size_t rem = i - (size_t)b * CC * NN;
  const float v = gamma[c] * (T[i] - mean[c]) * rsqrtf(var[c] + EPS_BN) + beta[c];
  const float r = X[i] + fmaxf(v, 0.f);
  XN[i] = r;
  Out[(size_t)b * 4 * CC * NN + rem] = r;   // slice base already offset by l*C*N
}

// ---------------------------------------------------------------------------
extern "C" void kernel_launch(void* const* d_in, const int* in_sizes, int n_in,
                              void* d_out, int out_size, void* d_ws, size_t ws_size,
                              hipStream_t stream) {
  (void)in_sizes; (void)n_in; (void)out_size; (void)ws_size;
  // setup_inputs() insertion order.
  const float* x   = (const float*)d_in[0];   // (B,C,N)
  const float* xyz = (const float*)d_in[1];   // (B,N,3)
  const float* Wc1 = (const float*)d_in[2];
  const float* Wc2 = (const float*)d_in[3];
  const float* g1  = (const float*)d_in[4];
  const float* be1 = (const float*)d_in[5];
  const float* g2  = (const float*)d_in[6];
  const float* be2 = (const float*)d_in[7];
  // SA layer leaves: Wqk, Wv, bv, Wt, bt, gamma, beta, W1, b1, W2, b2
  const int SA0 = 8, SAN = 11;

  const size_t SX = (size_t)BB * CC * NN;     // 1,048,576 floats
  float* ws    = (float*)d_ws;
  float* tbuf  = ws;                          // (B,C,N) GEMM scratch for BN
  float* xcur  = ws + 1 * SX;                 // current features
  float* xnext = ws + 2 * SX;                 // next features
  float* xv    = ws + 3 * SX;                 // value projection
  float* ubuf  = ws + 4 * SX;                 // x - x_r
  float* xqk   = ws + 5 * SX;                 // (B,64,N) = 262,144
  float* Pbuf  = xqk + (size_t)BB * C4 * NN;  // (B,N,16) = 65,536
  float* invb  = Pbuf + (size_t)BB * NN * PH; // (B,N)    = 4,096
  float* meanb = invb + (size_t)BB * NN;      // 256
  float* varb  = meanb + CC;                  // 256
  float* attn  = varb + CC;                   // (B,N,N)  = 4,194,304

  const dim3 blk32(32), blk128(128), blk256(256);
  const dim3 gFull(NN / 16, CC / 16, BB);     // O = 256 GEMM tiles
  const dim3 gQK(NN / 16, C4 / 16, BB);       // O = 64 GEMM tiles
  const dim3 gNN(NN / 16, NN / 16, BB);       // N x N score tiles
  const dim3 gXR(NN / 16, CC / 64, BB);       // 64x16 tiles, 4 waves/block

  // ---- conv1 + BN + relu ----
  k_gemm<<<gFull, blk32, 0, stream>>>(Wc1, x, nullptr, tbuf, CC);
  k_bn_stats<<<CC, blk256, 0, stream>>>(tbuf, meanb, varb);
  k_bn_relu<<<(int)(SX / 256), blk256, 0, stream>>>(tbuf, g1, be1, meanb, varb, xcur);
  // ---- conv2 + BN + relu ----
  k_gemm<<<gFull, blk32, 0, stream>>>(Wc2, xcur, nullptr, tbuf, CC);
  k_bn_stats<<<CC, blk256, 0, stream>>>(tbuf, meanb, varb);
  k_bn_relu<<<(int)(SX / 256), blk256, 0, stream>>>(tbuf, g2, be2, meanb, varb, xcur);

  // ---- 4 stacked SA layers ----
  for (int l = 0; l < 4; ++l) {
    const float* Wqk   = (const float*)d_in[SA0 + SAN * l + 0];
    const float* Wv    = (const float*)d_in[SA0 + SAN * l + 1];
    const float* bv    = (const float*)d_in[SA0 + SAN * l + 2];
    const float* Wt    = (const float*)d_in[SA0 + SAN * l + 3];
    const float* bt    = (const float*)d_in[SA0 + SAN * l + 4];
    const float* gamma = (const float*)d_in[SA0 + SAN * l + 5];
    const float* beta  = (const float*)d_in[SA0 + SAN * l + 6];
    const float* W1    = (const float*)d_in[SA0 + SAN * l + 7];
    const float* b1    = (const float*)d_in[SA0 + SAN * l + 8];
    const float* W2    = (const float*)d_in[SA0 + SAN * l + 9];
    const float* b2    = (const float*)d_in[SA0 + SAN * l + 10];

    k_gemm<<<gQK, blk32, 0, stream>>>(Wqk, xcur, nullptr, xqk, CC);      // Q/K (shared)
    k_gemm<<<gFull, blk32, 0, stream>>>(Wv, xcur, bv, xv, CC);           // V
    k_pos_embed<<<(BB * NN * PH) / 256, blk256, 0, stream>>>(xyz, W1, Pbuf);
    k_energy<<<gNN, blk32, 0, stream>>>(xqk, Pbuf, b1, W2, b2, attn);    // scores
    k_softmax<<<BB * NN, blk256, 0, stream>>>(attn);                     // row softmax
    k_colsum_inv<<<(BB * NN) / 256, blk256, 0, stream>>>(attn, invb);    // L1 col renorm
    k_xr_sub<<<gXR, blk128, 0, stream>>>(xv, attn, invb, xcur, ubuf);    // u = x - x_r
    k_gemm<<<gFull, blk32, 0, stream>>>(Wt, ubuf, bt, tbuf, CC);         // trans_conv
    k_bn_stats<<<CC, blk256, 0, stream>>>(tbuf, meanb, varb);
    k_residual<<<(int)(SX / 256), blk256, 0, stream>>>(
        xcur, tbuf, meanb, varb, gamma, beta, xnext,
        (float*)d_out + (size_t)l * CC * NN);

    float* tmp = xcur; xcur = xnext; xnext = tmp;
  }
}